// Net_53712861003991
// MI455X (gfx1250) — compile-verified
//
#include <hip/hip_runtime.h>

typedef __attribute__((ext_vector_type(2))) float v2f;
typedef __attribute__((ext_vector_type(8))) float v8f;

#define N_NODES 100000
#define F_IN    256
#define H1_DIM  128
#define H2_DIM  64
#define E_EDGES 1600000
#define EP_CNT  100000

// ---------------------------------------------------------------------------
// Degree / normalization
// ---------------------------------------------------------------------------
__global__ void init_deg_kernel(float* __restrict__ deg, int n) {
    int i = blockIdx.x * blockDim.x + threadIdx.x;
    if (i < n) deg[i] = 1.0f;  // self-loop contributes 1
}

__global__ void accum_deg_kernel(const int* __restrict__ col,
                                 float* __restrict__ deg, int e) {
    int i = blockIdx.x * blockDim.x + threadIdx.x;
    if (i < e) atomicAdd(&deg[col[i]], 1.0f);
}

__global__ void rsqrt_kernel(float* __restrict__ deg, int n) {
    int i = blockIdx.x * blockDim.x + threadIdx.x;
    if (i < n) deg[i] = rsqrtf(deg[i]);
}

// ---------------------------------------------------------------------------
// Weight transpose: B[K,N] -> Bt[N,K]  (tiny; amortized over 6250 GEMM blocks)
// ---------------------------------------------------------------------------
__global__ void transpose_kernel(const float* __restrict__ B,
                                 float* __restrict__ Bt, int K, int N) {
    int idx = blockIdx.x * blockDim.x + threadIdx.x;
    if (idx >= K * N) return;
    int k = idx / N;
    int n = idx - k * N;
    Bt[(size_t)n * K + k] = B[idx];
}

// ---------------------------------------------------------------------------
// GEMM: C[M, 16*NT] = A[M,K] @ Bt[16*NT, K]^T  (Bt is pre-transposed weights)
// fp32 WMMA 16x16x4, one wave per 16x16 output tile, K fully unrolled.
// A-fragment (32-bit A 16x4): lanes 0-15 hold M rows with K{0,1} in the VGPR
// pair; lanes 16-31 hold K{2,3}. With Bt transposed, the B fragment has the
// identical addressing pattern, so both fragments are single b64 loads at
// compile-time immediate offsets.
// ---------------------------------------------------------------------------
template <int K, int NT>
__global__ void __launch_bounds__(NT * 32)
gemm_wmma_f32(const float* __restrict__ A, const float* __restrict__ Bt,
              float* __restrict__ C) {
    constexpr int N = NT * 16;
    const int wave = threadIdx.x >> 5;
    const int lane = threadIdx.x & 31;
    const int l    = lane & 15;
    const int half = lane >> 4;          // 0: K pair {0,1}, 1: K pair {2,3}

    const int m0 = blockIdx.x * 16;
    const int n0 = wave * 16;

    const float* ap = A  + (size_t)(m0 + l) * K + 2 * half;
    const float* bp = Bt + (size_t)(n0 + l) * K + 2 * half;

    v8f acc = {0.f, 0.f, 0.f, 0.f, 0.f, 0.f, 0.f, 0.f};
#pragma unroll
    for (int k0 = 0; k0 < K; k0 += 4) {
        v2f a = *(const v2f*)(ap + k0);  // 8B-aligned: even element offset
        v2f b = *(const v2f*)(bp + k0);
        acc = __builtin_amdgcn_wmma_f32_16x16x4_f32(
            /*neg_a=*/false, a, /*neg_b=*/false, b,
            /*c_mod=*/(short)0, acc, /*reuse_a=*/false, /*reuse_b=*/false);
    }

    float* cp = C + (size_t)(m0 + 8 * half) * N + n0 + l;
#pragma unroll
    for (int j = 0; j < 8; ++j)
        cp[(size_t)j * N] = acc[j];
}

// ---------------------------------------------------------------------------
// Scatter phase
// ---------------------------------------------------------------------------
// acc[i,f] = htmp[i,f] * dinv[i]^2   (self-loop term; initializes accumulator)
template <int H>
__global__ void self_init_kernel(const float* __restrict__ htmp,
                                 const float* __restrict__ dinv,
                                 float* __restrict__ acc, int n) {
    int idx = blockIdx.x * blockDim.x + threadIdx.x;
    if (idx >= n * H) return;
    int node = idx / H;
    float di = dinv[node];
    acc[idx] = htmp[idx] * di * di;
}

// warp per edge: acc[c,:] += htmp[r,:] * dinv[r]*dinv[c]
template <int H>
__global__ void __launch_bounds__(256)
edge_scatter_kernel(const int* __restrict__ row, const int* __restrict__ col,
                    const float* __restrict__ dinv,
                    const float* __restrict__ htmp,
                    float* __restrict__ acc, int e) {
    int warp_id = (blockIdx.x * blockDim.x + threadIdx.x) >> 5;
    int lane = threadIdx.x & 31;
    if (warp_id >= e) return;
    int r = row[warp_id];
    int c = col[warp_id];
    float norm = dinv[r] * dinv[c];
    const float* src = htmp + (size_t)r * H;
    float* dst = acc + (size_t)c * H;
#pragma unroll
    for (int f = 0; f < H; f += 32)
        atomicAdd(&dst[f + lane], src[f + lane] * norm);
}

template <int H>
__global__ void bias_relu_kernel(float* __restrict__ acc,
                                 const float* __restrict__ bias, int n) {
    int idx = blockIdx.x * blockDim.x + threadIdx.x;
    if (idx >= n * H) return;
    float v = acc[idx] + bias[idx & (H - 1)];
    acc[idx] = v > 0.f ? v : 0.f;
}

template <int H>
__global__ void bias_add_kernel(float* __restrict__ acc,
                                const float* __restrict__ bias, int n) {
    int idx = blockIdx.x * blockDim.x + threadIdx.x;
    if (idx >= n * H) return;
    acc[idx] = acc[idx] + bias[idx & (H - 1)];
}

// ---------------------------------------------------------------------------
// Edge dot products: out[e] = <h2[te1[e]], h2[te0[e]]> over H2=64 features.
// Warp per edge, 2 features per lane, wave32 xor-shuffle reduction.
// ---------------------------------------------------------------------------
__global__ void __launch_bounds__(256)
edge_dot_kernel(const int* __restrict__ pos, const int* __restrict__ neg,
                const float* __restrict__ h2, float* __restrict__ out, int ep) {
    int warp_id = (blockIdx.x * blockDim.x + threadIdx.x) >> 5;
    int lane = threadIdx.x & 31;
    if (warp_id >= 2 * ep) return;
    int src, dst;
    if (warp_id < ep) {
        src = pos[warp_id];            // te[0] = source
        dst = pos[ep + warp_id];       // te[1] = target
    } else {
        int e2 = warp_id - ep;
        src = neg[e2];
        dst = neg[ep + e2];
    }
    const float* xj = h2 + (size_t)src * H2_DIM;
    const float* xi = h2 + (size_t)dst * H2_DIM;
    float s = xi[lane] * xj[lane] + xi[lane + 32] * xj[lane + 32];
#pragma unroll
    for (int off = 16; off > 0; off >>= 1)
        s += __shfl_xor(s, off, 32);
    if (lane == 0) out[warp_id] = s;
}

// ---------------------------------------------------------------------------
// Host launcher
// ---------------------------------------------------------------------------
static inline size_t align256(size_t x) { return (x + 255) & ~(size_t)255; }

extern "C" void kernel_launch(void* const* d_in, const int* in_sizes, int n_in,
                              void* d_out, int out_size, void* d_ws, size_t ws_size,
                              hipStream_t stream) {
    const float* x    = (const float*)d_in[0];
    const int*   ei   = (const int*)d_in[1];   // [2, E] flat
    const int*   pos  = (const int*)d_in[2];   // [2, EP] flat
    const int*   neg  = (const int*)d_in[3];   // [2, EP] flat
    const float* W1   = (const float*)d_in[4];
    const float* b1   = (const float*)d_in[5];
    const float* W2   = (const float*)d_in[6];
    const float* b2   = (const float*)d_in[7];
    float* out = (float*)d_out;

    const int* row = ei;            // edge_index[0]
    const int* col = ei + E_EDGES;  // edge_index[1]

    // Workspace carve-up (256-byte aligned regions)
    char* ws = (char*)d_ws;
    size_t off = 0;
    float* dinv  = (float*)(ws + off); off += align256(sizeof(float) * N_NODES);
    float* W1t   = (float*)(ws + off); off += align256(sizeof(float) * F_IN * H1_DIM);
    float* W2t   = (float*)(ws + off); off += align256(sizeof(float) * H1_DIM * H2_DIM);
    float* htmp1 = (float*)(ws + off); off += align256(sizeof(float) * (size_t)N_NODES * H1_DIM);
    float* h1    = (float*)(ws + off); off += align256(sizeof(float) * (size_t)N_NODES * H1_DIM);
    float* htmp2 = (float*)(ws + off); off += align256(sizeof(float) * (size_t)N_NODES * H2_DIM);
    float* h2    = (float*)(ws + off); off += align256(sizeof(float) * (size_t)N_NODES * H2_DIM);
    (void)ws_size; (void)n_in; (void)in_sizes; (void)out_size;

    const int TB = 256;

    // 0. Transpose weights for symmetric b64 fragment loads
    transpose_kernel<<<(F_IN * H1_DIM + TB - 1) / TB, TB, 0, stream>>>(W1, W1t, F_IN, H1_DIM);
    transpose_kernel<<<(H1_DIM * H2_DIM + TB - 1) / TB, TB, 0, stream>>>(W2, W2t, H1_DIM, H2_DIM);

    // 1. deg -> dinv
    init_deg_kernel<<<(N_NODES + TB - 1) / TB, TB, 0, stream>>>(dinv, N_NODES);
    accum_deg_kernel<<<(E_EDGES + TB - 1) / TB, TB, 0, stream>>>(col, dinv, E_EDGES);
    rsqrt_kernel<<<(N_NODES + TB - 1) / TB, TB, 0, stream>>>(dinv, N_NODES);

    // 2. GEMM1: htmp1 = x @ W1   (M=100000, K=256, 8 wave-tiles -> N=128)
    gemm_wmma_f32<F_IN, 8><<<N_NODES / 16, 8 * 32, 0, stream>>>(x, W1t, htmp1);

    // 3. Layer-1 scatter + bias + relu -> h1
    {
        int tot = N_NODES * H1_DIM;
        self_init_kernel<H1_DIM><<<(tot + TB - 1) / TB, TB, 0, stream>>>(htmp1, dinv, h1, N_NODES);
        int warps_blocks = (E_EDGES * 32 + TB - 1) / TB;
        edge_scatter_kernel<H1_DIM><<<warps_blocks, TB, 0, stream>>>(row, col, dinv, htmp1, h1, E_EDGES);
        bias_relu_kernel<H1_DIM><<<(tot + TB - 1) / TB, TB, 0, stream>>>(h1, b1, N_NODES);
    }

    // 4. GEMM2: htmp2 = h1 @ W2  (M=100000, K=128, 4 wave-tiles -> N=64)
    gemm_wmma_f32<H1_DIM, 4><<<N_NODES / 16, 4 * 32, 0, stream>>>(h1, W2t, htmp2);

    // 5. Layer-2 scatter + bias -> h2
    {
        int tot = N_NODES * H2_DIM;
        self_init_kernel<H2_DIM><<<(tot + TB - 1) / TB, TB, 0, stream>>>(htmp2, dinv, h2, N_NODES);
        int warps_blocks = (E_EDGES * 32 + TB - 1) / TB;
        edge_scatter_kernel<H2_DIM><<<warps_blocks, TB, 0, stream>>>(row, col, dinv, htmp2, h2, E_EDGES);
        bias_add_kernel<H2_DIM><<<(tot + TB - 1) / TB, TB, 0, stream>>>(h2, b2, N_NODES);
    }

    // 6. Edge dot products -> out (2*EP values)
    {
        long long total_threads = (long long)2 * EP_CNT * 32;
        int blocks = (int)((total_threads + TB - 1) / TB);
        edge_dot_kernel<<<blocks, TB, 0, stream>>>(pos, neg, h2, out, EP_CNT);
    }
}